// FuseSliceTogetherReplacement_7713761264256
// MI455X (gfx1250) — compile-verified
//
#include <hip/hip_runtime.h>

// Problem constants (match the reference exactly).
#define ROWS    4096   // input rows
#define COLS    1024   // input cols (row stride in floats)
#define NSLICE  256    // number of slices
#define SLEN    64     // floats per slice window
#define UNROLL  8      // rows per thread (stepped by 16)

typedef __attribute__((ext_vector_type(4))) float v4f;

// out[s, r, 0:64] = in[r, idx[s] : idx[s]+64]
//
// Layout per block (256 threads = 8 wave32 waves):
//   j4 = t & 15   -> which float4 within the 64-float row (lane-contiguous)
//   r0 = bx*128 + (t>>4), rows r0 + 16k for k = 0..7
//
// Per thread: 1 address setup, then 8x global_load_b128 (imm offsets k*65536B)
// and 8x global_store_b128 th:TH_STORE_NT (imm offsets k*4096B).
// idx[blockIdx.y] is block-uniform -> scalar s_load_b32.
__global__ __launch_bounds__(256) void
gather_slices_nt_kernel(const float* __restrict__ in,
                        const int*   __restrict__ idx,
                        float*       __restrict__ out)
{
    const int s  = blockIdx.y;                      // slice id (uniform)
    const int t  = threadIdx.x;                     // 0..255
    const int j4 = t & 15;                          // float4 within row
    const int r0 = blockIdx.x * (16 * UNROLL) + (t >> 4); // base row

    const int c0 = idx[s];                          // scalar load (uniform)

    const float* __restrict__ src =
        in + (size_t)r0 * COLS + (size_t)c0 + (size_t)(j4 << 2);
    float* __restrict__ dst =
        out + ((size_t)s * ROWS + (size_t)r0) * SLEN + (size_t)(j4 << 2);

    v4f v[UNROLL];
#pragma unroll
    for (int k = 0; k < UNROLL; ++k) {
        // Source only 4B-aligned (idx arbitrary) -> scalar loads; the
        // compiler merges them into one unaligned global_load_b128 with
        // immediate offset k*16*COLS*4 bytes.
        const float* __restrict__ p = src + (size_t)(k * 16 * COLS);
        v[k].x = p[0];
        v[k].y = p[1];
        v[k].z = p[2];
        v[k].w = p[3];
    }
#pragma unroll
    for (int k = 0; k < UNROLL; ++k) {
        // Destination 16B-aligned, write-once -> b128 NT store, immediate
        // offset k*16*SLEN*4 bytes; keeps the reused input resident in L2.
        __builtin_nontemporal_store(
            v[k], (v4f*)(dst + (size_t)(k * 16 * SLEN)));
    }
}

extern "C" void kernel_launch(void* const* d_in, const int* in_sizes, int n_in,
                              void* d_out, int out_size, void* d_ws, size_t ws_size,
                              hipStream_t stream)
{
    const float* in  = (const float*)d_in[0];  // (4096, 1024) f32
    const int*   idx = (const int*)  d_in[1];  // (256,) i32
    // d_in[2] is slice_len == 64 (compile-time constant here)
    float* out = (float*)d_out;                // (256, 4096, 64) f32

    dim3 grid(ROWS / (16 * UNROLL), NSLICE);   // (32, 256) = 8192 blocks
    dim3 block(256);                           // 8 wave32 waves
    gather_slices_nt_kernel<<<grid, block, 0, stream>>>(in, idx, out);
}